// GCNEncoder_61134564491387
// MI455X (gfx1250) — compile-verified
//
#include <hip/hip_runtime.h>
#include <hip/hip_bf16.h>

#define HID 128
#define LN_EPSF 1e-5f
#define KPAD 4
#define KMAX 256

typedef float v2f __attribute__((ext_vector_type(2)));
typedef float v8f __attribute__((ext_vector_type(8)));

// ------------------- degree / gcn_norm -------------------
__global__ void k_deg_init(float* __restrict__ deg, int n) {
  int i = blockIdx.x * blockDim.x + threadIdx.x;
  if (i < n) deg[i] = 1.0f;   // +1 self loop
}

__global__ void k_deg_acc(const long long* __restrict__ dst, float* __restrict__ deg, int E) {
  int e = blockIdx.x * blockDim.x + threadIdx.x;
  if (e < E) atomicAdd(&deg[(int)dst[e]], 1.0f);
}

__global__ void k_dinv(const float* __restrict__ deg, float* __restrict__ dinv, int n) {
  int i = blockIdx.x * blockDim.x + threadIdx.x;
  if (i < n) dinv[i] = rsqrtf(deg[i]);
}

// ------------------- WMMA f32 GEMM: C[M x 128] = A[M x K] @ B[K x 128] (+bias) ---------
// One block = 256 threads = 8 waves, one 16-row tile; wave w owns 16-col tile w.
// A tile (16 x K) is staged into LDS once per block via CDNA5 async global->LDS copies
// (ASYNCcnt), then all 8 waves feed V_WMMA_F32_16X16X4_F32 from conflict-free ds_load_b64.
__global__ __launch_bounds__(256) void k_gemm_f32_wmma(const float* __restrict__ A,
                                                       const float* __restrict__ B,
                                                       const float* __restrict__ bias,
                                                       float* __restrict__ C, int K) {
  __shared__ float sA[16 * (KMAX + KPAD)];
  const int KP  = K + KPAD;                 // padded row stride: 4m mod 64 distinct banks
  const int tid = threadIdx.x;
  const long long rowBase = (long long)blockIdx.x * 16;

  // ---- stage A tile into LDS: 16B chunks, uniform trip count (EXEC stays full) ----
  {
    const int k4      = K >> 2;             // 16B chunks per row
    const int nChunks = 16 * k4;            // 1024 (K=256) or 512 (K=128)
    const unsigned lds_base = (unsigned)(size_t)(&sA[0]);
    for (int c = tid; c < nChunks; c += 256) {
      const int r    = c / k4;
      const int col4 = (c - r * k4) << 2;
      const unsigned      ldsoff = lds_base + (unsigned)((r * KP + col4) * 4);
      const unsigned long long ga =
          (unsigned long long)(const void*)(A + (rowBase + r) * (long long)K + col4);
      asm volatile("global_load_async_to_lds_b128 %0, %1, off"
                   :: "v"(ldsoff), "v"(ga) : "memory");
    }
    asm volatile("s_wait_asynccnt 0x0" ::: "memory");
    __syncthreads();
  }

  const int lane = tid & 31;
  const int wave = tid >> 5;
  const int m    = lane & 15;
  const int hi   = lane >> 4;               // 0: K=0,1 ; 1: K=2,3 within each step
  const int nc   = wave * 16 + m;           // output column for B/C frags
  const float* __restrict__ arow = &sA[m * KP];

  v8f acc = {};
#pragma unroll 4
  for (int k0 = 0; k0 < K; k0 += 4) {
    const int ka = k0 + 2 * hi;
    const float2 av = *(const float2*)(arow + ka);  // ds_load_b64 (bank-conflict free)
    v2f a; a.x = av.x; a.y = av.y;
    v2f b; b.x = B[ka * HID + nc];                  // B[ka][nc]
    b.y = B[(ka + 1) * HID + nc];                   // B[ka+1][nc]
    acc = __builtin_amdgcn_wmma_f32_16x16x4_f32(false, a, false, b,
                                                (short)0, acc, false, false);
  }

  const float bv = bias ? bias[nc] : 0.0f;
#pragma unroll
  for (int r = 0; r < 8; ++r) {
    C[(rowBase + r + 8 * hi) * HID + nc] = acc[r] + bv;   // C[m=r+8*hi][nc]
  }
}

// ------------------- agg = hw * self_norm + conv_b[l] -------------------
__global__ void k_agg_init(const float* __restrict__ hw, const float* __restrict__ dinv,
                           const float* __restrict__ cb, float* __restrict__ agg, int n) {
  int t = blockIdx.x * blockDim.x + threadIdx.x;    // one thread = (node, 4 features)
  int i = t >> 5, lane = t & 31;
  if (i >= n) return;
  float sn = dinv[i]; sn *= sn;
  const long long off = (long long)i * HID + lane * 4;
  float4 v = *(const float4*)(hw + off);
  float4 b = *(const float4*)(cb + lane * 4);
  float4 o = { v.x * sn + b.x, v.y * sn + b.y, v.z * sn + b.z, v.w * sn + b.w };
  *(float4*)(agg + off) = o;
}

// ------------------- edge scatter: agg[dst] += hw[src] * dinv[src]*dinv[dst] ----------
__global__ void k_edge_scatter(const long long* __restrict__ src, const long long* __restrict__ dst,
                               const float* __restrict__ dinv, const float* __restrict__ hw,
                               float* __restrict__ agg, int E) {
  int t = blockIdx.x * blockDim.x + threadIdx.x;    // one wave = one edge (4 f32/lane)
  int e = t >> 5, lane = t & 31;
  if (e >= E) return;

  // prefetch the gathered row a few edges ahead (global_prefetch_b8)
  const int PF = 8;
  if (e + PF < E) {
    const int sp = (int)src[e + PF];
    __builtin_prefetch((const void*)(hw + (long long)sp * HID + lane * 4), 0, 0);
  }

  const int s = (int)src[e];
  const int d = (int)dst[e];
  const float w = dinv[s] * dinv[d];
  const float4 v = *(const float4*)(hw + (long long)s * HID + lane * 4);
  float* a = agg + (long long)d * HID + lane * 4;
  atomicAdd(a + 0, v.x * w);
  atomicAdd(a + 1, v.y * w);
  atomicAdd(a + 2, v.z * w);
  atomicAdd(a + 3, v.w * w);
}

// ------------------- fused LN -> ReLU -> residual -> JK accumulate -------------------
// One wave per node (32 lanes x 4 features = 128). Wave32 __shfl_xor reductions.
__global__ __launch_bounds__(256) void k_ln_relu_res_jk(const float* __restrict__ agg,
                                                        const float* __restrict__ g,
                                                        const float* __restrict__ bta,
                                                        const float* __restrict__ jk_w, int l,
                                                        float* __restrict__ h,
                                                        float* __restrict__ out,
                                                        int n, int first) {
  const int wave = threadIdx.x >> 5, lane = threadIdx.x & 31;
  const long long i = (long long)blockIdx.x * 8 + wave;
  if (i >= n) return;
  const float jkw = jk_w[l];
  const long long off = i * HID + lane * 4;

  float4 v = *(const float4*)(agg + off);
  float s = v.x + v.y + v.z + v.w;
#pragma unroll
  for (int o = 16; o >= 1; o >>= 1) s += __shfl_xor(s, o, 32);
  const float mu = s * (1.0f / HID);

  float4 d = { v.x - mu, v.y - mu, v.z - mu, v.w - mu };
  float q = d.x * d.x + d.y * d.y + d.z * d.z + d.w * d.w;
#pragma unroll
  for (int o = 16; o >= 1; o >>= 1) q += __shfl_xor(q, o, 32);
  const float inv = rsqrtf(q * (1.0f / HID) + LN_EPSF);

  const float4 gg = *(const float4*)(g + lane * 4);
  const float4 bb = *(const float4*)(bta + lane * 4);
  const float4 hp = *(const float4*)(h + off);
  float4 hn;
  hn.x = fmaxf(d.x * inv * gg.x + bb.x, 0.0f) + hp.x;
  hn.y = fmaxf(d.y * inv * gg.y + bb.y, 0.0f) + hp.y;
  hn.z = fmaxf(d.z * inv * gg.z + bb.z, 0.0f) + hp.z;
  hn.w = fmaxf(d.w * inv * gg.w + bb.w, 0.0f) + hp.w;
  *(float4*)(h + off) = hn;

  float4 ov;
  if (first) {
    ov.x = jkw * hn.x; ov.y = jkw * hn.y; ov.z = jkw * hn.z; ov.w = jkw * hn.w;
  } else {
    float4 op = *(const float4*)(out + off);
    ov.x = op.x + jkw * hn.x; ov.y = op.y + jkw * hn.y;
    ov.z = op.z + jkw * hn.z; ov.w = op.w + jkw * hn.w;
  }
  *(float4*)(out + off) = ov;
}

extern "C" void kernel_launch(void* const* d_in, const int* in_sizes, int n_in,
                              void* d_out, int out_size, void* d_ws, size_t ws_size,
                              hipStream_t stream) {
  const float*     x      = (const float*)d_in[0];
  const long long* ei     = (const long long*)d_in[1];   // int64 [2, E]
  const float*     W_in   = (const float*)d_in[2];
  const float*     b_in   = (const float*)d_in[3];
  const float*     conv_W = (const float*)d_in[4];       // [L, H, H]
  const float*     conv_b = (const float*)d_in[5];       // [L, H]
  const float*     ln_g   = (const float*)d_in[6];
  const float*     ln_b   = (const float*)d_in[7];
  const float*     jk_w   = (const float*)d_in[8];       // [L]

  const int in_ch = in_sizes[2] / HID;                   // 256
  const int n     = in_sizes[0] / in_ch;                 // 50000
  const int E     = in_sizes[1] / 2;                     // 800000
  const int L     = in_sizes[8];                         // 3
  const long long NH = (long long)n * HID;

  float* ws   = (float*)d_ws;
  float* deg  = ws;                // n
  float* dinv = deg + n;           // n
  float* h    = dinv + n;          // n*H
  float* hw   = h + NH;            // n*H
  float* agg  = hw + NH;           // n*H
  float* out  = (float*)d_out;

  const long long* src = ei;
  const long long* dst = ei + E;

  // gcn_norm
  k_deg_init<<<(n + 255) / 256, 256, 0, stream>>>(deg, n);
  k_deg_acc<<<(E + 255) / 256, 256, 0, stream>>>(dst, deg, E);
  k_dinv<<<(n + 255) / 256, 256, 0, stream>>>(deg, dinv, n);

  // h = x @ W_in + b_in   (WMMA f32, K = in_ch)
  k_gemm_f32_wmma<<<n / 16, 256, 0, stream>>>(x, W_in, b_in, h, in_ch);

  for (int l = 0; l < L; ++l) {
    // hw = h @ conv_W[l]
    k_gemm_f32_wmma<<<n / 16, 256, 0, stream>>>(h, conv_W + (long long)l * HID * HID,
                                                nullptr, hw, HID);
    // agg = hw * self_norm + conv_b[l]
    {
      long long threads = (long long)n * 32;
      k_agg_init<<<(unsigned)((threads + 255) / 256), 256, 0, stream>>>(
          hw, dinv, conv_b + (long long)l * HID, agg, n);
    }
    // agg[dst] += hw[src] * enorm
    {
      long long threads = (long long)E * 32;
      k_edge_scatter<<<(unsigned)((threads + 255) / 256), 256, 0, stream>>>(
          src, dst, dinv, hw, agg, E);
    }
    // h = relu(LN(agg)) + h ; out (+)= jk_w[l] * h
    k_ln_relu_res_jk<<<(n + 7) / 8, 256, 0, stream>>>(
        agg, ln_g + (long long)l * HID, ln_b + (long long)l * HID, jk_w, l,
        h, out, n, l == 0);
  }
}